// ConvNet2_56521769615919
// MI455X (gfx1250) — compile-verified
//
#include <hip/hip_runtime.h>
#include <math.h>

typedef float v2f __attribute__((ext_vector_type(2)));
typedef float v8f __attribute__((ext_vector_type(8)));

#define ENC_NEG_INF 0x007FFFFFu   // encode(bits(-inf)) = ~0xFF800000

__device__ __forceinline__ unsigned fenc(float f) {
  unsigned b = __float_as_uint(f);
  return (b & 0x80000000u) ? ~b : (b | 0x80000000u);
}
__device__ __forceinline__ float fdec(unsigned k) {
  return (k & 0x80000000u) ? __uint_as_float(k ^ 0x80000000u) : __uint_as_float(~k);
}

// ---------------- fill ----------------
__global__ void fill_u32(unsigned* __restrict__ p, unsigned v, int n) {
  int i = blockIdx.x * blockDim.x + threadIdx.x;
  if (i < n) p[i] = v;
}

// ---------------- degree count ----------------
__global__ void deg_count(const int* __restrict__ dst, int* __restrict__ deg, int E) {
  int e = blockIdx.x * blockDim.x + threadIdx.x;
  if (e < E) atomicAdd(&deg[dst[e]], 1);
}

// ---------------- layer-0 scalar scatter (add, F=1) ----------------
__global__ void scatter_add_f1(const int* __restrict__ src, const int* __restrict__ dst,
                               const float* __restrict__ x, float* __restrict__ agg, int E) {
  int e = blockIdx.x * blockDim.x + threadIdx.x;
  if (e < E) atomicAdd(&agg[dst[e]], x[src[e]]);
}

// ---------------- F=32 scatter: wave per edge, lane = feature ----------------
__global__ void scatter_add32(const int* __restrict__ src, const int* __restrict__ dst,
                              const float* __restrict__ x, float* __restrict__ agg, int E) {
  int lane = threadIdx.x & 31;
  int wave = (blockIdx.x * blockDim.x + threadIdx.x) >> 5;
  int nw   = (gridDim.x * blockDim.x) >> 5;
  for (int e = wave; e < E; e += nw) {
    int s = src[e], d = dst[e];
    atomicAdd(&agg[d * 32 + lane], x[s * 32 + lane]);
  }
}

__global__ void scatter_max32(const int* __restrict__ src, const int* __restrict__ dst,
                              const float* __restrict__ x, unsigned* __restrict__ aggu, int E) {
  int lane = threadIdx.x & 31;
  int wave = (blockIdx.x * blockDim.x + threadIdx.x) >> 5;
  int nw   = (gridDim.x * blockDim.x) >> 5;
  for (int e = wave; e < E; e += nw) {
    int s = src[e], d = dst[e];
    atomicMax(&aggu[d * 32 + lane], fenc(x[s * 32 + lane]));
  }
}

// ---------------- layer 0 update (1 -> 32), relu ----------------
__global__ void update0(const float* __restrict__ agg, const float* __restrict__ x,
                        const float* __restrict__ Wr, const float* __restrict__ br,
                        const float* __restrict__ Wt, float* __restrict__ out, int nNodes) {
  int idx = blockIdx.x * blockDim.x + threadIdx.x;
  if (idx >= nNodes * 32) return;
  int n = idx >> 5, f = idx & 31;
  float v = agg[n] * Wr[f] + br[f] + x[n] * Wt[f];
  out[idx] = v > 0.f ? v : 0.f;
}

// ---------------- 32->32 dense update via V_WMMA_F32_16X16X4_F32 ----------------
// Each wave computes a 16-node x 32-feature tile:
//   out = aggr(agg) @ Wr + br + x @ Wt, then ReLU.
// mode: 0=add, 1=mean (scale by 1/max(deg,1)), 2=max (decode ordered-uint, sentinel->0)
//
// A (16x4 f32) layout per ISA: lane L holds row m=L&15; lanes 0-15 -> K={0,1},
// lanes 16-31 -> K={2,3} in (v0,v1). B (4x16) mirrored: lane L holds col n=L&15,
// v_r holds row k = (L>>4)*2 + r. C/D: vgpr r, lane L -> row m = r + (L>>4)*8, col n=L&15.
__global__ __launch_bounds__(256) void gconv32_wmma(
    const float* __restrict__ x, const float* __restrict__ aggf,
    const unsigned* __restrict__ aggu, const int* __restrict__ deg,
    const float* __restrict__ Wr, const float* __restrict__ br,
    const float* __restrict__ Wt, float* __restrict__ out, int nNodes, int mode) {
  int wave = threadIdx.x >> 5;
  int lane = threadIdx.x & 31;
  int tile = blockIdx.x * (blockDim.x >> 5) + wave;
  if (tile * 16 >= nNodes) return;        // whole-wave uniform exit; EXEC stays all-ones

  int m    = lane & 15;
  int kg   = (lane >> 4) << 1;            // 0 or 2
  int node = tile * 16 + m;

  float scale = 1.0f;
  if (mode == 1) {
    int dg = deg[node];
    scale = 1.0f / (float)(dg > 1 ? dg : 1);
  }

  v2f aA[8], aX[8];
  const float* xrow = x + node * 32;
#pragma unroll
  for (int ks = 0; ks < 8; ++ks) {
    int k0 = ks * 4 + kg;
    float a0, a1;
    if (mode == 2) {
      unsigned u0 = aggu[node * 32 + k0];
      unsigned u1 = aggu[node * 32 + k0 + 1];
      a0 = (u0 == ENC_NEG_INF) ? 0.f : fdec(u0);
      a1 = (u1 == ENC_NEG_INF) ? 0.f : fdec(u1);
    } else {
      a0 = aggf[node * 32 + k0] * scale;
      a1 = aggf[node * 32 + k0 + 1] * scale;
    }
    v2f ta = {a0, a1};
    aA[ks] = ta;
    v2f tx = {xrow[k0], xrow[k0 + 1]};
    aX[ks] = tx;
  }

  int rowbase = tile * 16 + ((lane >> 4) << 3);
#pragma unroll
  for (int nt = 0; nt < 2; ++nt) {
    int n = nt * 16 + m;                  // output column handled by this lane
    v8f c = {};
#pragma unroll
    for (int ks = 0; ks < 8; ++ks) {
      int r0 = ks * 4 + kg;
      v2f bR = {Wr[r0 * 32 + n], Wr[(r0 + 1) * 32 + n]};
      c = __builtin_amdgcn_wmma_f32_16x16x4_f32(false, aA[ks], false, bR,
                                                (short)0, c, false, false);
      v2f bT = {Wt[r0 * 32 + n], Wt[(r0 + 1) * 32 + n]};
      c = __builtin_amdgcn_wmma_f32_16x16x4_f32(false, aX[ks], false, bT,
                                                (short)0, c, false, false);
    }
    float bias = br[n];
#pragma unroll
    for (int r = 0; r < 8; ++r) {
      float v = c[r] + bias;
      v = v > 0.f ? v : 0.f;
      out[(rowbase + r) * 32 + n] = v;
    }
  }
}

// ---------------- final layer (32 -> 1), max aggr, sigmoid ----------------
__global__ void final_layer(const float* __restrict__ x, const unsigned* __restrict__ aggu,
                            const float* __restrict__ Wr, const float* __restrict__ br,
                            const float* __restrict__ Wt, float* __restrict__ out, int nNodes) {
  int lane = threadIdx.x & 31;
  int node = (blockIdx.x * blockDim.x + threadIdx.x) >> 5;
  if (node >= nNodes) return;
  unsigned u = aggu[node * 32 + lane];
  float a = (u == ENC_NEG_INF) ? 0.f : fdec(u);
  float v = a * Wr[lane] + x[node * 32 + lane] * Wt[lane];
#pragma unroll
  for (int off = 16; off > 0; off >>= 1) v += __shfl_down(v, off, 32);
  if (lane == 0) out[node] = 1.f / (1.f + expf(-(v + br[0])));
}

extern "C" void kernel_launch(void* const* d_in, const int* in_sizes, int n_in,
                              void* d_out, int out_size, void* d_ws, size_t ws_size,
                              hipStream_t stream) {
  const float* x0 = (const float*)d_in[0];
  const int N = in_sizes[0];
  const int E = in_sizes[1] / 2;
  const int* src = (const int*)d_in[1];
  const int* dst = src + E;
  const float *Wr[6], *br[6], *Wt[6];
  for (int i = 0; i < 6; ++i) {
    Wr[i] = (const float*)d_in[2 + 3 * i];
    br[i] = (const float*)d_in[3 + 3 * i];
    Wt[i] = (const float*)d_in[4 + 3 * i];
  }

  // workspace: cur[N*32] | nxt[N*32] | agg[N*32] | deg[N]  (~38.8 MB)
  float*    cur = (float*)d_ws;
  float*    nxt = cur + (size_t)N * 32;
  unsigned* agg = (unsigned*)(nxt + (size_t)N * 32);
  int*      deg = (int*)(agg + (size_t)N * 32);

  const int T = 256;
  auto nb = [](long n, int t) { return (int)((n + t - 1) / t); };
  const int NF = N * 32;
  const int SC_BLOCKS = 4096;  // 32768 waves looping over edges

  // degrees (shared by both mean layers)
  fill_u32<<<nb(N, T), T, 0, stream>>>((unsigned*)deg, 0u, N);
  deg_count<<<nb(E, T), T, 0, stream>>>(dst, deg, E);

  // ---- layer 0: add, 1 -> 32 ----
  fill_u32<<<nb(N, T), T, 0, stream>>>(agg, 0u, N);
  scatter_add_f1<<<nb(E, T), T, 0, stream>>>(src, dst, x0, (float*)agg, E);
  update0<<<nb(NF, T), T, 0, stream>>>((const float*)agg, x0, Wr[0], br[0], Wt[0], cur, N);

  // ---- layers 1..4: 32 -> 32 (mean, add, mean, max) ----
  int tiles = (N + 15) / 16;
  int gblocks = nb((long)tiles * 32, T);
  for (int i = 1; i <= 4; ++i) {
    int mode = (i == 1 || i == 3) ? 1 : (i == 4 ? 2 : 0);
    if (mode == 2) {
      fill_u32<<<nb(NF, T), T, 0, stream>>>(agg, ENC_NEG_INF, NF);
      scatter_max32<<<SC_BLOCKS, T, 0, stream>>>(src, dst, cur, agg, E);
    } else {
      fill_u32<<<nb(NF, T), T, 0, stream>>>(agg, 0u, NF);
      scatter_add32<<<SC_BLOCKS, T, 0, stream>>>(src, dst, cur, (float*)agg, E);
    }
    gconv32_wmma<<<gblocks, T, 0, stream>>>(cur, (const float*)agg, agg, deg,
                                            Wr[i], br[i], Wt[i], nxt, N, mode);
    float* t = cur; cur = nxt; nxt = t;
  }

  // ---- layer 5: max, 32 -> 1, sigmoid ----
  fill_u32<<<nb(NF, T), T, 0, stream>>>(agg, ENC_NEG_INF, NF);
  scatter_max32<<<SC_BLOCKS, T, 0, stream>>>(src, dst, cur, agg, E);
  final_layer<<<nb(NF, T), T, 0, stream>>>(cur, agg, Wr[5], br[5], Wt[5], (float*)d_out, N);
}